// LinearCRF_44616120271059
// MI455X (gfx1250) — compile-verified
//
#include <hip/hip_runtime.h>
#include <stdint.h>

// Viterbi decode (linear-chain CRF), B=256, S=2048, T=64, on gfx1250.
// Max-plus recursion cannot use WMMA ((+,x) semiring only). CDNA5-specific
// paths: async global->LDS feature streaming (ASYNCcnt / s_wait_asynccnt),
// wave32 cross-lane argmax split, LDS-broadcast b128 score reads.

#define NEGV (-10000.0f)
constexpr int Bn = 256, Sn = 2048, Tn = 64;
constexpr int START_T = 61, STOP_T = 62, PAD_T = 63;

// ---- async global->LDS copy of one b32 per lane (gfx1250 async-tensor path) ----
__device__ __forceinline__ void async_b32_to_lds(void* lds_dst, const float* gbase,
                                                 uint32_t lane_byte_off) {
#if defined(__HIP_DEVICE_COMPILE__)
  // Flat shared pointer low 32 bits == LDS byte offset (ISA 10.2 aperture rule).
  uint32_t loff  = (uint32_t)(uintptr_t)lds_dst;
  uint64_t sbase = (uint64_t)(uintptr_t)gbase;
  asm volatile("global_load_async_to_lds_b32 %0, %1, %2"
               :: "v"(loff), "v"(lane_byte_off), "s"(sbase)
               : "memory");
#else
  (void)lds_dst; (void)gbase; (void)lane_byte_off;
#endif
}

__device__ __forceinline__ void wait_async0() {
#if defined(__HIP_DEVICE_COMPILE__)
  asm volatile("s_wait_asynccnt 0" ::: "memory");
#endif
}

// ---------------- forward: max-plus recursion + backpointers ----------------
// 1 block per batch, 128 threads = 4 wave32. Lane pair (2*j, 2*j+1) owns tag j;
// each half reduces 32 predecessor tags from register-resident transitions.
__global__ __launch_bounds__(128, 1)
void viterbi_fwd(const float* __restrict__ features,
                 const float* __restrict__ mask,
                 const float* __restrict__ transitions,
                 uint8_t* __restrict__ bptr,
                 int* __restrict__ best_tag,
                 int* __restrict__ lens,
                 float* __restrict__ best_score)
{
  const int b    = blockIdx.x;
  const int tid  = threadIdx.x;   // 0..127
  const int j    = tid >> 1;      // tag 0..63
  const int half = tid & 1;       // which half of predecessors (i in [32h,32h+32))

  __shared__ __align__(16) float s_score[2][Tn];
  __shared__ __align__(16) float s_feat[2][Tn];
  __shared__ float s_red[Tn];
  __shared__ int   s_len;

  // ---- sequence length = sum(mask[b,:]) : 4x b128 loads + wave32 shfl tree ----
  {
    const float4* mp = (const float4*)(mask + (size_t)b * Sn);
    float part = 0.f;
    #pragma unroll
    for (int k = 0; k < 4; ++k) {
      float4 v = mp[tid + 128 * k];
      part += (v.x + v.y) + (v.z + v.w);
    }
    #pragma unroll
    for (int d = 16; d >= 1; d >>= 1) part += __shfl_xor(part, d, 32);
    if ((tid & 31) == 0) s_red[tid >> 5] = part;  // one partial per wave
  }
  __syncthreads();
  if (tid == 0) s_len = (int)(s_red[0] + s_red[1] + s_red[2] + s_red[3] + 0.5f);

  // ---- transitions[j, 32*half .. 32*half+31] into 32 VGPRs ----
  float tr[32];
  {
    const float4* tp = (const float4*)(transitions + (size_t)j * Tn + half * 32);
    #pragma unroll
    for (int k = 0; k < 8; ++k) {
      float4 v = tp[k];
      tr[4*k+0] = v.x; tr[4*k+1] = v.y; tr[4*k+2] = v.z; tr[4*k+3] = v.w;
    }
  }

  // ---- init score, async-prefetch features[b,0,:] into LDS ----
  if (half == 0) s_score[0][j] = (j == START_T) ? 0.f : NEGV;
  if (tid < Tn) {
    async_b32_to_lds(&s_feat[0][tid], features + (size_t)b * Sn * Tn,
                     (uint32_t)tid * 4u);
  }
  wait_async0();
  __syncthreads();

  const int L = s_len;            // 1024 <= L <= 2048
  const int ibase = half * 32;
  float cur_score = s_score[0][j];

  // hoisted, incremented pointers (kills per-step 64-bit s_mul chains)
  const float* gnext = features + ((size_t)b * Sn + 1) * Tn;  // row t+1
  uint8_t*     bpp   = bptr + (size_t)b * Sn * Tn;            // row t

  for (int t = 0; t < L; ++t) {
    const int cur = t & 1, nxt = cur ^ 1;

    // prefetch next feature row while computing this step
    if ((t + 1 < L) && (tid < Tn)) {
      async_b32_to_lds(&s_feat[nxt][tid], gnext, (uint32_t)tid * 4u);
    }

    // max-plus reduce over this lane's 32 predecessors, 4 independent strands
    float m0 = -INFINITY, m1 = -INFINITY, m2 = -INFINITY, m3 = -INFINITY;
    int   a0 = 0, a1 = 1, a2 = 2, a3 = 3;
    #pragma unroll
    for (int q = 0; q < 8; ++q) {
      const float4 sv = *(const float4*)&s_score[cur][ibase + 4*q]; // LDS broadcast b128
      const int i0 = ibase + 4*q;
      float c0 = sv.x + tr[4*q+0]; if (c0 > m0) { m0 = c0; a0 = i0;     }
      float c1 = sv.y + tr[4*q+1]; if (c1 > m1) { m1 = c1; a1 = i0 + 1; }
      float c2 = sv.z + tr[4*q+2]; if (c2 > m2) { m2 = c2; a2 = i0 + 2; }
      float c3 = sv.w + tr[4*q+3]; if (c3 > m3) { m3 = c3; a3 = i0 + 3; }
    }
    // merge strands, exact first-argmax tie rule (needed: NEG-filled transition
    // rows/cols create exact ties; smaller index must win, like jnp.argmax)
    float m = m0; int a = a0;
    if (m1 > m || (m1 == m && a1 < a)) { m = m1; a = a1; }
    if (m2 > m || (m2 == m && a2 < a)) { m = m2; a = a2; }
    if (m3 > m || (m3 == m && a3 < a)) { m = m3; a = a3; }
    // cross-half merge with partner lane (wave32 shuffle)
    float om = __shfl_xor(m, 1, 32);
    int   oa = __shfl_xor(a, 1, 32);
    if (om > m || (om == m && oa < a)) { m = om; a = oa; }

    // score update (t < L always inside this loop, so mask==1)
    float ns = m + s_feat[cur][j];
    cur_score = ns;
    if (half == 0) {
      s_score[nxt][j] = ns;
      bpp[j] = (uint8_t)a;
    }
    gnext += Tn;
    bpp   += Tn;

    wait_async0();      // next feature row resident before barrier
    __syncthreads();
  }

  // ---- final = score + transitions[STOP,:]; block argmax (first max) ----
  float fin = cur_score + transitions[(size_t)STOP_T * Tn + j];
  if (half == 0) s_red[j] = fin;
  __syncthreads();
  if (tid == 0) {
    float bm = s_red[0]; int ba = 0;
    for (int k = 1; k < Tn; ++k) { float v = s_red[k]; if (v > bm) { bm = v; ba = k; } }
    best_tag[b]   = ba;
    lens[b]       = L;
    best_score[b] = bm;
  }
}

// ---------------- backtrace: chunked pointer-chase through LDS ----------------
constexpr int CH = 128;  // steps per LDS chunk (8 KB of backpointers)

__global__ __launch_bounds__(64, 1)
void viterbi_back(const uint8_t* __restrict__ bptr,
                  const int* __restrict__ best_tag,
                  const int* __restrict__ lens,
                  int* __restrict__ path)
{
  const int b   = blockIdx.x;
  const int tid = threadIdx.x; // 0..63
  __shared__ uint8_t s_bp[CH][Tn];
  __shared__ int     s_path[CH];

  const int L  = lens[b];
  const int bt = best_tag[b];
  int carry = bt;              // only thread 0's copy is meaningful

  for (int cs = Sn - CH; cs >= 0; cs -= CH) {
    // coalesced chunk load of backpointers into LDS
    const uint32_t* src = (const uint32_t*)(bptr + ((size_t)b * Sn + cs) * Tn);
    uint32_t* dst = (uint32_t*)&s_bp[0][0];
    #pragma unroll
    for (int k = tid; k < CH * Tn / 4; k += 64) dst[k] = src[k];
    __syncthreads();

    if (tid == 0) {
      int c = carry;
      for (int tt = CH - 1; tt >= 0; --tt) {
        const int t = cs + tt;
        if (t >= L) { s_path[tt] = PAD_T; continue; }  // carry reset at L-1 makes skipped updates irrelevant
        if (t == L - 1) c = bt;
        s_path[tt] = c;
        c = s_bp[tt][c];       // LDS-latency chain instead of L2 pointer-chase
      }
      carry = c;
    }
    __syncthreads();

    #pragma unroll
    for (int tt = tid; tt < CH; tt += 64) path[(size_t)b * Sn + cs + tt] = s_path[tt];
    __syncthreads();
  }
}

extern "C" void kernel_launch(void* const* d_in, const int* in_sizes, int n_in,
                              void* d_out, int out_size, void* d_ws, size_t ws_size,
                              hipStream_t stream) {
  (void)in_sizes; (void)n_in; (void)out_size; (void)ws_size;
  const float* features    = (const float*)d_in[0];   // [B,S,T] f32
  const float* mask        = (const float*)d_in[1];   // [B,S]   f32
  const float* transitions = (const float*)d_in[2];   // [T,T]   f32

  int*   path       = (int*)d_out;                                  // [B,S] int32
  float* best_score = (float*)((int*)d_out + (size_t)Bn * Sn);      // [B]   f32

  uint8_t* ws       = (uint8_t*)d_ws;
  uint8_t* bptr     = ws;                                           // B*S*T bytes = 32 MiB
  int*     best_tag = (int*)(ws + (size_t)Bn * Sn * Tn);
  int*     lens     = best_tag + Bn;

  viterbi_fwd <<<Bn, 128, 0, stream>>>(features, mask, transitions,
                                       bptr, best_tag, lens, best_score);
  viterbi_back<<<Bn,  64, 0, stream>>>(bptr, best_tag, lens, path);
}